// GNNEncoder_3478923510413
// MI455X (gfx1250) — compile-verified
//
#include <hip/hip_runtime.h>
#include <hip/hip_bf16.h>

typedef __attribute__((ext_vector_type(2))) float v2f;
typedef __attribute__((ext_vector_type(8))) float v8f;

#define IN_DIM 256
#define HID_DIM 64

#define A_PAD 260   // words per A-tile row in LDS (even: b64-aligned; conflict-free reads)
#define W_PAD 262   // words per W^T row in LDS (even: b64-aligned; conflict-free)
#define SMEM_WORDS (HID_DIM * W_PAD + 16 * A_PAD)

// ---------------- degree kernels ----------------
__global__ void __launch_bounds__(256) init_deg_kernel(float* __restrict__ deg, int n) {
    int i = blockIdx.x * blockDim.x + threadIdx.x;
    if (i < n) deg[i] = 1.0f;   // self-loop contributes 1 to every node's degree
}

__global__ void __launch_bounds__(256) count_deg_kernel(const long long* __restrict__ cols,
                                                        float* __restrict__ deg, int nE) {
    int e = blockIdx.x * blockDim.x + threadIdx.x;
    if (e < nE) {
        int t = (int)cols[e];
        __hip_atomic_fetch_add(&deg[t], 1.0f, __ATOMIC_RELAXED, __HIP_MEMORY_SCOPE_AGENT);
    }
}

__global__ void __launch_bounds__(256) dinv_kernel(float* __restrict__ deg, int n) {
    int i = blockIdx.x * blockDim.x + threadIdx.x;
    if (i < n) deg[i] = __frsqrt_rn(deg[i]);   // deg >= 1 always (self loop)
}

// ---------------- WMMA GEMM: g = (x @ W) * dinv[row]; out seeded with g -------------
// Persistent blocks (grid-stride over 16-row M tiles). Block = 128 threads = 4 waves,
// one per 16-col N tile of HID=64. W^T staged once per block in LDS; A tile staged
// per M tile. Inner loop: 2x ds_load_b64 + 1x V_WMMA_F32_16X16X4_F32.
__global__ void __launch_bounds__(128) gemm_g_kernel(const float* __restrict__ x,
                                                     const float* __restrict__ W,
                                                     const float* __restrict__ dinv,
                                                     float* __restrict__ g,
                                                     float* __restrict__ out,
                                                     int mtiles) {
    extern __shared__ float smem[];
    float* lds_wt = smem;                     // wt[n][k], n in [0,64), row pad W_PAD
    float* lds_a  = smem + HID_DIM * W_PAD;   // a[m][k],  m in [0,16), row pad A_PAD

    const int tid  = threadIdx.x;
    const int wave = tid >> 5;                // 0..3 -> N tile
    const int lane = tid & 31;
    const int lm   = lane & 15;               // position within 16
    const int hi   = lane >> 4;               // lane half
    const int koff = hi << 1;                 // K sub-offset: 0 or 2
    const int n0   = wave << 4;

    // Stage W transposed into LDS once (coalesced global b32 reads).
    for (int idx = tid; idx < IN_DIM * HID_DIM; idx += 128) {
        const int k = idx >> 6;               // W is [256][64] row-major
        const int n = idx & 63;
        lds_wt[n * W_PAD + k] = W[idx];
    }
    __syncthreads();

    const float* wt = lds_wt + (n0 + lm) * W_PAD;   // this lane's B column strip

    for (int mt = blockIdx.x; mt < mtiles; mt += gridDim.x) {
        const int m0 = mt << 4;

        // Stage 16x256 A tile (coalesced float2 global reads -> padded LDS rows).
        {
            const float2* xs = (const float2*)(x + (size_t)m0 * IN_DIM);
            for (int idx = tid; idx < 16 * (IN_DIM / 2); idx += 128) {
                const int r = idx >> 7;       // IN_DIM/2 = 128 float2 per row
                const int c = idx & 127;
                const float2 v = xs[(size_t)r * 128 + c];
                *(float2*)(lds_a + r * A_PAD + (c << 1)) = v;
            }
        }
        __syncthreads();

        const float* arow = lds_a + lm * A_PAD;
        v8f c = {};
#pragma unroll 8
        for (int k = 0; k < IN_DIM; k += 4) {
            // A 16x4 f32: lane holds M=lm, K = koff, koff+1 (contiguous b64 in LDS)
            v2f a = *(const v2f*)(arow + k + koff);
            // B 4x16 f32: lane holds N=n0+lm, K = koff, koff+1 (contiguous in W^T)
            v2f b = *(const v2f*)(wt + k + koff);
            c = __builtin_amdgcn_wmma_f32_16x16x4_f32(
                    /*neg_a=*/false, a, /*neg_b=*/false, b,
                    /*c_mod=*/(short)0, c, /*reuse_a=*/false, /*reuse_b=*/false);
        }

        // Epilogue: D VGPR r holds row m = 8*hi + r, col n = n0 + lm.
        const int mbase = m0 + (hi << 3);
#pragma unroll
        for (int r = 0; r < 8; ++r) {
            const int m = mbase + r;
            const float val = c[r] * dinv[m];       // g = h * dinv (row-normalized)
            const size_t idx = (size_t)m * HID_DIM + n0 + lm;
            g[idx]   = val;                          // gather source for edges
            out[idx] = val;                          // self-loop seed of accumulator
        }
        __syncthreads();   // protect lds_a (WAR) before next tile's staging
    }
}

// ---------------- edge scatter: out[t] += g[s] ----------------
// 16 threads per edge, 4 floats each (float4 gather + 4 L2-resident atomics).
__global__ void __launch_bounds__(256) scatter_kernel(const long long* __restrict__ rows,
                                                      const long long* __restrict__ cols,
                                                      const float* __restrict__ g,
                                                      float* __restrict__ out, int nE) {
    int gid = blockIdx.x * blockDim.x + threadIdx.x;
    int e = gid >> 4;
    if (e >= nE) return;
    const int chunk = (gid & 15) << 2;          // 0,4,...,60
    const int s = (int)rows[e];
    const int t = (int)cols[e];
    const float4 gv = *(const float4*)(g + (size_t)s * HID_DIM + chunk);
    float* dst = out + (size_t)t * HID_DIM + chunk;
    __hip_atomic_fetch_add(dst + 0, gv.x, __ATOMIC_RELAXED, __HIP_MEMORY_SCOPE_AGENT);
    __hip_atomic_fetch_add(dst + 1, gv.y, __ATOMIC_RELAXED, __HIP_MEMORY_SCOPE_AGENT);
    __hip_atomic_fetch_add(dst + 2, gv.z, __ATOMIC_RELAXED, __HIP_MEMORY_SCOPE_AGENT);
    __hip_atomic_fetch_add(dst + 3, gv.w, __ATOMIC_RELAXED, __HIP_MEMORY_SCOPE_AGENT);
}

// ---------------- finish: out = out * dinv[t] + b ----------------
__global__ void __launch_bounds__(256) finish_kernel(float* __restrict__ out,
                                                     const float* __restrict__ dinv,
                                                     const float* __restrict__ b, int total) {
    int i = blockIdx.x * blockDim.x + threadIdx.x;
    if (i < total) {
        const int t = i >> 6;                   // node
        const int j = i & 63;                   // feature
        out[i] = out[i] * dinv[t] + b[j];
    }
}

extern "C" void kernel_launch(void* const* d_in, const int* in_sizes, int n_in,
                              void* d_out, int out_size, void* d_ws, size_t ws_size,
                              hipStream_t stream) {
    const float*     x  = (const float*)d_in[0];      // [N, 256]
    const long long* ei = (const long long*)d_in[1];  // [2, E] int64
    const float*     W  = (const float*)d_in[2];      // [256, 64]
    const float*     b  = (const float*)d_in[3];      // [64]

    const int N = in_sizes[0] / IN_DIM;
    const int E = in_sizes[1] / 2;
    const long long* rows = ei;       // edge_index[0] = sources
    const long long* cols = ei + E;   // edge_index[1] = targets

    float* deg = (float*)d_ws;                       // N floats; becomes dinv in place
    float* g   = deg + (((size_t)N + 255) & ~(size_t)255);  // N*64 floats
    float* out = (float*)d_out;

    // 1. degree (with self loops) -> dinv
    init_deg_kernel<<<(N + 255) / 256, 256, 0, stream>>>(deg, N);
    count_deg_kernel<<<(E + 255) / 256, 256, 0, stream>>>(cols, deg, E);
    dinv_kernel<<<(N + 255) / 256, 256, 0, stream>>>(deg, N);

    // 2. WMMA GEMM: g = (x@W)*dinv[row], out seeded with g (self loop term).
    //    Persistent blocks; W^T staged once per block (LDS ~84KB -> 3 blocks/WGP).
    const int mtiles = (N + 15) / 16;                // 6250, exact
    const int gblocks = (mtiles < 1536) ? mtiles : 1536;
    gemm_g_kernel<<<gblocks, 128, SMEM_WORDS * sizeof(float), stream>>>(x, W, deg, g, out, mtiles);

    // 3. edge scatter-add (L2-resident fp32 atomics)
    {
        long long threads = (long long)E * 16;
        int blocks = (int)((threads + 255) / 256);
        scatter_kernel<<<blocks, 256, 0, stream>>>(rows, cols, g, out, E);
    }

    // 4. scale by dinv[target] and add bias
    finish_kernel<<<(N * HID_DIM + 255) / 256, 256, 0, stream>>>(out, deg, b, N * HID_DIM);
}